// UmbrellaSurfaceConstructor_26104811225238
// MI455X (gfx1250) — compile-verified
//
#include <hip/hip_runtime.h>
#include <math.h>

typedef __attribute__((ext_vector_type(2))) float v2f;
typedef __attribute__((ext_vector_type(8))) float v8f;

#define NPTS 4096
#define NBATCH 4
#define KNN 8
#define BIGF 3.0e38f
#define PI_F 3.14159265358979323846f

// ---------------------------------------------------------------------------
// Kernel A: fused pairwise-distance (WMMA f32 16x16x4) + per-query top-9.
// Block = 256 threads = 8 waves; each wave owns 16 query points; all waves
// share one LDS copy of the precomputed candidate A-operand [-2x,-2y,-2z,sq]
// (4096 x 4 f32 = 64 KB of the 320 KB WGP LDS).
// D[m][q] = sq_m - 2*dot(q,m)  (rank-equivalent to the true distance; the
// +sq_q term is constant per query column and cannot change the top-k).
// C-layout (wave32): lane q (q<16) holds candidates 0..7 in d[0..7], lane
// q+16 holds candidates 8..15 -> per-lane top-9, then shfl_xor(16) merge.
// ---------------------------------------------------------------------------
__global__ __launch_bounds__(256) void knn_wmma_kernel(const float* __restrict__ center,
                                                       int* __restrict__ nbr)
{
    __shared__ float sA[NPTS * 4];        // 64 KB: [-2x,-2y,-2z,sq] per point

    const int lane = threadIdx.x & 31;
    const int wave = threadIdx.x >> 5;
    const int tile = blockIdx.x * 8 + wave;     // 0..(NBATCH*NPTS/16 - 1)
    const int b = tile >> 8;                    // NPTS/16 == 256 tiles per batch
    const int qbase = (tile & 255) << 4;
    const int lq = lane & 15;
    const bool hi = lane >= 16;
    const int q = qbase + lq;
    const float* cb = center + (size_t)b * NPTS * 3;

    // cooperative stage: one global read + transform per point per block
    {
        float4* sA4 = (float4*)sA;
        for (int p = threadIdx.x; p < NPTS; p += 256) {
            const float x = cb[3 * p + 0];
            const float y = cb[3 * p + 1];
            const float z = cb[3 * p + 2];
            const float sq = x * x + y * y + z * z;
            sA4[p] = make_float4(-2.0f * x, -2.0f * y, -2.0f * z, sq);
        }
    }
    __syncthreads();

    // B matrix (queries): lanes 0-15 hold (K0,K1)=(x,y); lanes 16-31 (K2,K3)=(z,1)
    const float qx = cb[3 * q + 0], qy = cb[3 * q + 1], qz = cb[3 * q + 2];
    v2f bmat;
    bmat.x = hi ? qz : qx;
    bmat.y = hi ? 1.0f : qy;

    float bd[9]; int bi[9];
#pragma unroll
    for (int j = 0; j < 9; ++j) { bd[j] = BIGF; bi[j] = 0x7fffffff; }

    const int aoff = hi ? 2 : 0;  // lanes 0-15: (K0,K1); lanes 16-31: (K2,K3)

#pragma unroll 2
    for (int m0 = 0; m0 < NPTS; m0 += 16) {
        const int m = m0 + lq;
        // A operand straight from LDS: ds_load_b64, zero VALU prep
        const v2f amat = *(const v2f*)&sA[4 * m + aoff];
        v8f cz = {0.f, 0.f, 0.f, 0.f, 0.f, 0.f, 0.f, 0.f};
        v8f d = __builtin_amdgcn_wmma_f32_16x16x4_f32(false, amat, false, bmat,
                                                      (short)0, cz, false, false);
        // single screen per tile: min of the 8 candidate distances in this lane
        float t01 = fminf(d[0], d[1]), t23 = fminf(d[2], d[3]);
        float t45 = fminf(d[4], d[5]), t67 = fminf(d[6], d[7]);
        const float vmin = fminf(fminf(t01, t23), fminf(t45, t67));
        if (vmin < bd[8]) {                       // rare path
            const int base = m0 + (hi ? 8 : 0);
#pragma unroll
            for (int v = 0; v < 8; ++v) {
                const float dv = d[v];
                const int idx = base + v;
                if (dv < bd[8]) {
                    bd[8] = dv; bi[8] = idx;
#pragma unroll
                    for (int j = 8; j > 0; --j) {   // single bubble pass
                        const bool sw = bd[j] < bd[j - 1];
                        const float tf = bd[j - 1]; const int ti = bi[j - 1];
                        bd[j - 1] = sw ? bd[j] : bd[j - 1];
                        bi[j - 1] = sw ? bi[j] : bi[j - 1];
                        bd[j] = sw ? tf : bd[j];
                        bi[j] = sw ? ti : bi[j];
                    }
                }
            }
        }
    }

    // merge lane l's 9-list with lane (l^16)'s 9-list -> smallest 9 of 18
    float od[9]; int oi[9];
#pragma unroll
    for (int j = 0; j < 9; ++j) {
        od[j] = __shfl_xor(bd[j], 16, 32);
        oi[j] = __shfl_xor(bi[j], 16, 32);
    }
    int ri[9];
#pragma unroll
    for (int k = 0; k < 9; ++k) {
        const bool ta = (bd[0] < od[0]) || ((bd[0] == od[0]) && (bi[0] < oi[0]));
        ri[k] = ta ? bi[0] : oi[0];
#pragma unroll
        for (int j = 0; j < 8; ++j) {
            bd[j] = ta ? bd[j + 1] : bd[j];  bi[j] = ta ? bi[j + 1] : bi[j];
            od[j] = ta ? od[j] : od[j + 1];  oi[j] = ta ? oi[j] : oi[j + 1];
        }
        bd[8] = ta ? BIGF : bd[8];
        od[8] = ta ? od[8] : BIGF;
    }
    if (!hi) {
        int* o = nbr + ((size_t)b * NPTS + q) * KNN;
#pragma unroll
        for (int j = 0; j < KNN; ++j) o[j] = ri[j + 1];   // drop self (rank 0)
    }
}

// ---------------------------------------------------------------------------
// Kernel B: per-point umbrella geometry + BN-folded 3-layer MLP + sum over 8.
// ---------------------------------------------------------------------------
__global__ __launch_bounds__(256) void umbrella_mlp_kernel(
    const float* __restrict__ center, const int* __restrict__ nbr,
    const float* __restrict__ w1, const float* __restrict__ g1, const float* __restrict__ be1,
    const float* __restrict__ m1, const float* __restrict__ v1,
    const float* __restrict__ w2, const float* __restrict__ b2, const float* __restrict__ g2,
    const float* __restrict__ be2, const float* __restrict__ m2, const float* __restrict__ v2,
    const float* __restrict__ w3, const float* __restrict__ b3,
    float* __restrict__ out)
{
    const int t = blockIdx.x * 256 + threadIdx.x;   // 0..NBATCH*NPTS-1
    const int b = t / NPTS, n = t % NPTS;
    const float* cb = center + (size_t)b * NPTS * 3;
    const float cx0 = cb[3 * n + 0], cy0 = cb[3 * n + 1], cz0 = cb[3 * n + 2];

    float gx[8], gy[8], gz[8], ph[8];
    const int* nb = nbr + (size_t)t * KNN;
#pragma unroll
    for (int j = 0; j < 8; ++j) {
        const int m = nb[j];
        gx[j] = cb[3 * m + 0] - cx0;
        gy[j] = cb[3 * m + 1] - cy0;
        gz[j] = cb[3 * m + 2] - cz0;
        float xa = gx[j]; if (fabsf(xa) < 1e-10f) xa += 1e-10f;
        float ya = gy[j]; if (fabsf(ya) < 1e-10f) ya += 1e-10f;
        ph[j] = atan2f(ya, xa);                 // monotone in reference phi
    }

    // 19-comparator sorting network on (ph, gx, gy, gz)
#define CSWAP(a, c)                                                            \
    do {                                                                       \
        if (ph[a] > ph[c]) {                                                   \
            float t0;                                                          \
            t0 = ph[a]; ph[a] = ph[c]; ph[c] = t0;                             \
            t0 = gx[a]; gx[a] = gx[c]; gx[c] = t0;                             \
            t0 = gy[a]; gy[a] = gy[c]; gy[c] = t0;                             \
            t0 = gz[a]; gz[a] = gz[c]; gz[c] = t0;                             \
        }                                                                      \
    } while (0)
    CSWAP(0,1); CSWAP(2,3); CSWAP(4,5); CSWAP(6,7);
    CSWAP(0,2); CSWAP(1,3); CSWAP(4,6); CSWAP(5,7);
    CSWAP(1,2); CSWAP(5,6); CSWAP(0,4); CSWAP(3,7);
    CSWAP(1,5); CSWAP(2,6);
    CSWAP(1,4); CSWAP(3,6);
    CSWAP(2,4); CSWAP(3,5);
    CSWAP(3,4);
#undef CSWAP

    // triangles (0, s_j, s_{j+1}): normals + centers
    float nx[8], ny[8], nz[8], ccx[8], ccy[8], ccz[8];
#pragma unroll
    for (int j = 0; j < 8; ++j) {
        const int j2 = (j + 1) & 7;
        const float ax = gx[j], ay = gy[j], az = gz[j];
        const float bx = gx[j2], by = gy[j2], bz = gz[j2];
        const float cxv = ay * bz - az * by;
        const float cyv = az * bx - ax * bz;
        const float czv = ax * by - ay * bx;
        const float nr = sqrtf(cxv * cxv + cyv * cyv + czv * czv);
        const float dv = (nr < 1e-6f) ? 1.0f : nr;
        nx[j] = cxv / dv; ny[j] = cyv / dv; nz[j] = czv / dv;
        ccx[j] = (ax + bx) * (1.0f / 3.0f);
        ccy[j] = (ay + by) * (1.0f / 3.0f);
        ccz[j] = (az + bz) * (1.0f / 3.0f);
    }
    const float pos = (nx[0] > 0.0f) ? 1.0f : -1.0f;

    float ff[8][10];
    bool msk[8];
#pragma unroll
    for (int j = 0; j < 8; ++j) {
        nx[j] *= pos; ny[j] *= pos; nz[j] *= pos;
        const float cx = ccx[j], cy = ccy[j], cz = ccz[j];
        const float rho = sqrtf(cx * cx + cy * cy + cz * cz + 1e-10f);
        float rt = cz / fmaxf(rho, 1e-10f);
        rt = fminf(fmaxf(rt, -1.0f + 1e-10f), 1.0f - 1e-10f);
        const float theta = acosf(rt) * (1.0f / PI_F);
        float px = cx; if (fabsf(px) < 1e-10f) px += 1e-10f;
        float py = cy; if (fabsf(py) < 1e-10f) py += 1e-10f;
        const float phi = atan2f(py, px) * (1.0f / (2.0f * PI_F)) + 0.5f;
        const float nn2 = sqrtf(nx[j] * nx[j] + ny[j] * ny[j] + nz[j] * nz[j]);
        float snx, sny, snz;
        if (nn2 < 1e-6f) { snx = 1.0f; sny = 0.0f; snz = 0.0f; }
        else { const float dd = fmaxf(nn2, 1e-6f); snx = nx[j] / dd; sny = ny[j] / dd; snz = nz[j] / dd; }
        const float scx = fminf(fmaxf(cx, -1e6f), 1e6f);
        const float scy = fminf(fmaxf(cy, -1e6f), 1e6f);
        const float scz = fminf(fmaxf(cz, -1e6f), 1e6f);
        const float gpos = (snx * scx + sny * scy + snz * scz) / (sqrtf(3.0f) + 1e-6f);
        msk[j] = (nx[j] != nx[j]) || (ny[j] != ny[j]) || (nz[j] != nz[j]);
        ff[j][0] = cx;  ff[j][1] = cy;  ff[j][2] = cz;
        ff[j][3] = rho; ff[j][4] = theta; ff[j][5] = phi;
        ff[j][6] = nx[j]; ff[j][7] = ny[j]; ff[j][8] = nz[j];
        ff[j][9] = gpos;
    }

    // _fix_nan: replace masked entries (center, normal, pos channels) by first unmasked
    {
        int first = 0; bool found = false;
#pragma unroll
        for (int j = 0; j < 8; ++j) {
            const bool take = (!found) && (!msk[j]);
            first = take ? j : first;
            found = found || (!msk[j]);
        }
        float fv0 = 0, fv1 = 0, fv2 = 0, fv6 = 0, fv7 = 0, fv8 = 0, fv9 = 0;
#pragma unroll
        for (int j = 0; j < 8; ++j) {
            const bool p = (j == first);
            fv0 = p ? ff[j][0] : fv0; fv1 = p ? ff[j][1] : fv1; fv2 = p ? ff[j][2] : fv2;
            fv6 = p ? ff[j][6] : fv6; fv7 = p ? ff[j][7] : fv7; fv8 = p ? ff[j][8] : fv8;
            fv9 = p ? ff[j][9] : fv9;
        }
#pragma unroll
        for (int j = 0; j < 8; ++j) {
            if (msk[j]) {
                ff[j][0] = fv0; ff[j][1] = fv1; ff[j][2] = fv2;
                ff[j][6] = fv6; ff[j][7] = fv7; ff[j][8] = fv8; ff[j][9] = fv9;
            }
        }
    }

    // BN folding
    float inv1[10], off1[10], inv2[10], off2[10];
#pragma unroll
    for (int c = 0; c < 10; ++c) {
        const float iv1 = g1[c] * rsqrtf(v1[c] + 1e-5f);
        inv1[c] = iv1; off1[c] = be1[c] - m1[c] * iv1;
        const float iv2 = g2[c] * rsqrtf(v2[c] + 1e-5f);
        inv2[c] = iv2; off2[c] = be2[c] + (b2[c] - m2[c]) * iv2;
    }

    float h1[8][10];
#pragma unroll
    for (int o = 0; o < 10; ++o) {
        float w[10];
#pragma unroll
        for (int i = 0; i < 10; ++i) w[i] = w1[o * 10 + i];
#pragma unroll
        for (int j = 0; j < 8; ++j) {
            float s = 0.0f;
#pragma unroll
            for (int i = 0; i < 10; ++i) s += w[i] * ff[j][i];
            h1[j][o] = fmaxf(s * inv1[o] + off1[o], 0.0f);
        }
    }
    float h2[8][10];
#pragma unroll
    for (int o = 0; o < 10; ++o) {
        float w[10];
#pragma unroll
        for (int i = 0; i < 10; ++i) w[i] = w2[o * 10 + i];
#pragma unroll
        for (int j = 0; j < 8; ++j) {
            float s = 0.0f;
#pragma unroll
            for (int i = 0; i < 10; ++i) s += w[i] * h1[j][i];
            h2[j][o] = fmaxf(s * inv2[o] + off2[o], 0.0f);
        }
    }
#pragma unroll
    for (int o = 0; o < 10; ++o) {
        float w[10];
#pragma unroll
        for (int i = 0; i < 10; ++i) w[i] = w3[o * 10 + i];
        float acc = 8.0f * b3[o];
#pragma unroll
        for (int j = 0; j < 8; ++j) {
            float s = 0.0f;
#pragma unroll
            for (int i = 0; i < 10; ++i) s += w[i] * h2[j][i];
            acc += s;
        }
        out[((size_t)b * 10 + o) * NPTS + n] = acc;
    }
}

extern "C" void kernel_launch(void* const* d_in, const int* in_sizes, int n_in,
                              void* d_out, int out_size, void* d_ws, size_t ws_size,
                              hipStream_t stream)
{
    const float* center = (const float*)d_in[0];
    const float* w1  = (const float*)d_in[1];
    const float* g1  = (const float*)d_in[2];
    const float* be1 = (const float*)d_in[3];
    const float* m1  = (const float*)d_in[4];
    const float* v1  = (const float*)d_in[5];
    const float* w2  = (const float*)d_in[6];
    const float* b2  = (const float*)d_in[7];
    const float* g2  = (const float*)d_in[8];
    const float* be2 = (const float*)d_in[9];
    const float* m2  = (const float*)d_in[10];
    const float* v2  = (const float*)d_in[11];
    const float* w3  = (const float*)d_in[12];
    const float* b3  = (const float*)d_in[13];
    int* nbr = (int*)d_ws;   // NBATCH*NPTS*KNN int32 = 512 KB

    knn_wmma_kernel<<<dim3(NBATCH * NPTS / 16 / 8), dim3(256), 0, stream>>>(center, nbr);
    umbrella_mlp_kernel<<<dim3(NBATCH * NPTS / 256), dim3(256), 0, stream>>>(
        center, nbr, w1, g1, be1, m1, v1, w2, b2, g2, be2, m2, v2, w3, b3, (float*)d_out);
}